// SpikingNeuralNetworkMock_41317585387921
// MI455X (gfx1250) — compile-verified
//
#include <hip/hip_runtime.h>
#include <hip/hip_bf16.h>

typedef __attribute__((ext_vector_type(2))) float v2f;
typedef __attribute__((ext_vector_type(8))) float v8f;

// C[M,N] = A[M,K] @ W[N,K]^T + bias[N], one wave32 per 16x16 tile,
// K-loop of V_WMMA_F32_16X16X4_F32.
__global__ void gemm_bias_wmma(const float* __restrict__ A,
                               const float* __restrict__ W,
                               const float* __restrict__ bias,
                               float* __restrict__ C,
                               int M, int N, int K) {
    const int lane  = threadIdx.x & 31;
    const int gwave = blockIdx.x * (blockDim.x >> 5) + (threadIdx.x >> 5);
    const int tilesN = N >> 4;
    const int tileM  = gwave / tilesN;
    const int tileN  = gwave - tileM * tilesN;
    if (tileM * 16 >= M) return;          // wave-uniform: EXEC stays all-1s

    const int half = lane >> 4;           // 0: K=k,k+1  1: K=k+2,k+3
    const int l15  = lane & 15;

    const float* __restrict__ Arow = A + (size_t)(tileM * 16 + l15) * K;
    const float* __restrict__ Wrow = W + (size_t)(tileN * 16 + l15) * K;

    v8f acc = {};
    for (int k = 0; k < K; k += 4) {
        const int kk = k + (half << 1);   // even -> 8B aligned float2 loads
        const float2 av = *(const float2*)(Arow + kk);
        const float2 wv = *(const float2*)(Wrow + kk);
        v2f a; a.x = av.x; a.y = av.y;    // A-frag: row=l15, K=kk,kk+1
        v2f b; b.x = wv.x; b.y = wv.y;    // B-frag (=W^T): col=l15, K=kk,kk+1
        acc = __builtin_amdgcn_wmma_f32_16x16x4_f32(
            /*neg_a=*/false, a, /*neg_b=*/false, b,
            /*c_mod=*/(short)0, acc, /*reuse_a=*/false, /*reuse_b=*/false);
    }

    const float bv = bias[tileN * 16 + l15];
    // C/D layout: VGPR v -> row v + 8*half, col l15
    float* Cp = C + (size_t)(tileM * 16 + (half << 3)) * N + (tileN * 16 + l15);
#pragma unroll
    for (int v = 0; v < 8; ++v) {
        Cp[(size_t)v * N] = acc[v] + bv;
    }
}

// One thread per (b, n): 50-step LIF recurrence in registers,
// spikes[b,n,0:50] stored as 25x float2 (contiguous 200B/thread),
// running mean for the next layer, optional second spike copy.
template <int T>
__global__ void lif_kernel(const float* __restrict__ I,
                           float* __restrict__ spikesA,
                           float* __restrict__ spikesB,   // may be nullptr
                           float* __restrict__ mean,      // may be nullptr
                           int total) {
    const int idx = blockIdx.x * blockDim.x + threadIdx.x;
    if (idx >= total) return;

    const float cur = I[idx];
    float V = 0.0f;
    float acc = 0.0f;
    float sp[T];
#pragma unroll
    for (int t = 0; t < T; ++t) {
        V = V * 0.9f + cur;                      // V = V*DECAY + I
        const float s = (V > 1.0f) ? 1.0f : 0.0f;
        V = V - V * s;                           // V *= (1 - s)
        sp[t] = s;
        acc += s;
    }

    const size_t base = (size_t)idx * T;         // [B, out, T] layout
#pragma unroll
    for (int t = 0; t < T; t += 2) {
        *(float2*)(spikesA + base + t) = make_float2(sp[t], sp[t + 1]);
    }
    if (spikesB) {
#pragma unroll
        for (int t = 0; t < T; t += 2) {
            *(float2*)(spikesB + base + t) = make_float2(sp[t], sp[t + 1]);
        }
    }
    if (mean) mean[idx] = acc * (1.0f / (float)T);
}

extern "C" void kernel_launch(void* const* d_in, const int* in_sizes, int n_in,
                              void* d_out, int out_size, void* d_ws, size_t ws_size,
                              hipStream_t stream) {
    const float* x  = (const float*)d_in[0];
    const float* W0 = (const float*)d_in[1];
    const float* b0 = (const float*)d_in[2];
    const float* W1 = (const float*)d_in[3];
    const float* b1 = (const float*)d_in[4];
    const float* W2 = (const float*)d_in[5];
    const float* b2 = (const float*)d_in[6];

    const int H    = in_sizes[2];            // 2048
    const int Din  = in_sizes[1] / H;        // 1024
    const int Dout = in_sizes[6];            // 1024
    const int B    = in_sizes[0] / Din;      // 512
    constexpr int T = 50;                    // time_steps (setup_inputs)

    float* out = (float*)d_out;
    // return order: (s2, s0, s1, s2)
    float* s2a = out;
    float* s0  = s2a + (size_t)B * Dout * T;
    float* s1  = s0  + (size_t)B * H * T;
    float* s2b = s1  + (size_t)B * H * T;

    // workspace: I (B*H floats) | mean (B*H floats)  => 8 MB
    float* I    = (float*)d_ws;
    float* mean = I + (size_t)B * H;

    auto gemm = [&](const float* Ain, const float* Wm, const float* bv,
                    float* Cout, int M, int N, int K) {
        const int tiles  = (M / 16) * (N / 16);
        const int blocks = (tiles + 7) / 8;       // 8 waves (tiles) per block
        gemm_bias_wmma<<<blocks, 256, 0, stream>>>(Ain, Wm, bv, Cout, M, N, K);
    };

    // layer 0: I0 = x @ W0^T + b0 ; LIF -> s0, mean0
    gemm(x, W0, b0, I, B, H, Din);
    {
        const int total = B * H;
        lif_kernel<T><<<(total + 255) / 256, 256, 0, stream>>>(I, s0, nullptr, mean, total);
    }
    // layer 1: I1 = mean0 @ W1^T + b1 ; LIF -> s1, mean1
    gemm(mean, W1, b1, I, B, H, H);
    {
        const int total = B * H;
        lif_kernel<T><<<(total + 255) / 256, 256, 0, stream>>>(I, s1, nullptr, mean, total);
    }
    // layer 2: I2 = mean1 @ W2^T + b2 ; LIF -> s2 (both copies)
    gemm(mean, W2, b2, I, B, Dout, H);
    {
        const int total = B * Dout;
        lif_kernel<T><<<(total + 255) / 256, 256, 0, stream>>>(I, s2a, s2b, nullptr, total);
    }
}